// SignatureTransform_41205916238128
// MI455X (gfx1250) — compile-verified
//
#include <hip/hip_runtime.h>

// Depth-4 path signature (Chen scan), DIM=8, streaming all prefixes.
// 3-phase associative scan: chunk totals -> WMMA-based sequential combine ->
// chunk replay with streamed float4 output writes.

#define DIMP   8
#define D2     64
#define D3     512
#define D4     4096
#define SIG    4680          // 8 + 64 + 512 + 4096
#define SIGQ   1170          // SIG / 4 (float4 count)
#define OFF1   0
#define OFF2   8
#define OFF3   72
#define OFF4   584
#define CLEN   64            // increments per chunk (CLEN*DIMP == TPB)
#define TPB    512

typedef float v2f __attribute__((ext_vector_type(2)));
typedef float v8f __attribute__((ext_vector_type(8)));
typedef int   b128v __attribute__((vector_size(4 * sizeof(int))));  // builtin's pointee type

// ---------------------------------------------------------------------------
// CDNA5 async global->LDS DMA (ASYNCcnt-tracked), with plain-load fallback.
// Param types per hipcc diagnostic: (int4 AS1*, int4 AS3*, imm offset, imm cpol).
// ---------------------------------------------------------------------------
#if defined(__AMDGCN__) && __has_builtin(__builtin_amdgcn_global_load_async_to_lds_b128)
#define ASYNC_LDS 1
__device__ __forceinline__ void async_ld_b128(const float* g, float* l)
{
    b128v* gg = (b128v*)(void*)(g);     // generic typed pointers first...
    b128v* ll = (b128v*)(void*)(l);
    __builtin_amdgcn_global_load_async_to_lds_b128(
        (__attribute__((address_space(1))) b128v*)gg,   // ...then pure AS conversion
        (__attribute__((address_space(3))) b128v*)ll,
        0, 0);
}
__device__ __forceinline__ void async_wait0()
{
#if __has_builtin(__builtin_amdgcn_s_wait_asynccnt)
    __builtin_amdgcn_s_wait_asynccnt(0);
#else
    asm volatile("s_wait_asynccnt 0" ::: "memory");
#endif
}
#endif

// ---------------------------------------------------------------------------
// Phase 1 / Phase 3: fold exp(z_t) into the carry with Chen's relation.
// b = exp(z) is rank-1, so:
//   c4'[m,l] = c4 + z_l * ( c3[m] + 0.5*c2[ij]*z_k + (c1[i]+z_i/4)*z_j*z_k/6 )
//   c3'[ijk] = c3 + c2[ij]*z_k + 0.5*(c1[i]+z_i/3)*z_j*z_k
//   c2'[ij]  = c2 + (c1[i]+z_i/2)*z_j
//   c1'      = c1 + z
// Thread tid owns c3[tid] and c4[tid*8..tid*8+7] privately in registers.
// ---------------------------------------------------------------------------
__global__ __launch_bounds__(TPB)
void sig_chunk_kernel(const float* __restrict__ path,
                      const float* __restrict__ init,   // exclusive prefixes (phase 3), null for phase 1
                      float* __restrict__ dst,          // chunk totals (phase 1) or streamed output (phase 3)
                      int nsteps, int phase)
{
    __shared__ float zc[CLEN * DIMP];   // staged increments for the chunk
    __shared__ float s1[DIMP];
    __shared__ float s2[D2];

    const int tid = threadIdx.x;
    const int c   = blockIdx.x;
    const int t0  = c * CLEN;
    int nst = nsteps - t0;
    if (nst > CLEN) nst = CLEN;
    if (nst < 0)    nst = 0;

#ifdef ASYNC_LDS
    __shared__ float zraw[(CLEN + 1) * DIMP];           // raw path rows t0 .. t0+nst
    {
        // (nst+1) rows * 8 floats = (nst+1)*2 float4s  (max 130, one pass)
        const int q4 = tid;
        if (q4 < (nst + 1) * 2)
            async_ld_b128(path + (size_t)t0 * DIMP + q4 * 4, zraw + q4 * 4);
    }
#else
    {
        const int s = tid >> 3, l = tid & 7;
        const int t = t0 + s;
        float v = 0.f;
        if (t < nsteps)
            v = path[(size_t)(t + 1) * DIMP + l] - path[(size_t)t * DIMP + l];
        zc[tid] = v;
    }
#endif

    float c4r[8];
    float c3r;
    if (phase == 0) {
#pragma unroll
        for (int l = 0; l < 8; ++l) c4r[l] = 0.f;
        c3r = 0.f;
        if (tid < D2)   s2[tid] = 0.f;
        if (tid < DIMP) s1[tid] = 0.f;
    } else {
        const float* Pc = init + (size_t)c * SIG;
#pragma unroll
        for (int l = 0; l < 8; ++l) c4r[l] = Pc[OFF4 + tid * 8 + l];
        c3r = Pc[OFF3 + tid];
        if (tid < D2)   s2[tid] = Pc[OFF2 + tid];
        if (tid < DIMP) s1[tid] = Pc[OFF1 + tid];
    }

#ifdef ASYNC_LDS
    async_wait0();
    __syncthreads();
    {
        const int s = tid >> 3;
        zc[tid] = (s < nst) ? (zraw[tid + DIMP] - zraw[tid]) : 0.f;
    }
#endif

    const int ii = tid >> 6;            // m = (i,j,k) flattened, m == tid
    const int jj = (tid >> 3) & 7;
    const int kk = tid & 7;
    __syncthreads();

    for (int s = 0; s < nst; ++s) {
        const float* z = &zc[s * 8];
        // ---- read phase (old state) ----
        float zr[8];
#pragma unroll
        for (int l = 0; l < 8; ++l) zr[l] = z[l];
        const float zi = z[ii], zj = z[jj], zk = z[kk];
        const float c2m = s2[tid >> 3];
        const float c1m = s1[ii];
        float c2own = 0.f, c1q = 0.f, c1own = 0.f;
        if (tid < D2)   { c2own = s2[tid]; c1q = s1[tid >> 3]; }
        if (tid < DIMP)   c1own = s1[tid];

        // ---- compute (registers only) ----
        const float zjk = zj * zk;
        const float g   = c3r + 0.5f * c2m * zk + (1.f / 6.f) * (c1m + 0.25f * zi) * zjk;
        const float n3  = c3r + c2m * zk + 0.5f * (c1m + (1.f / 3.f) * zi) * zjk;
#pragma unroll
        for (int l = 0; l < 8; ++l) c4r[l] += g * zr[l];
        c3r = n3;
        float n2 = 0.f, n1 = 0.f;
        if (tid < D2)   n2 = c2own + (c1q + 0.5f * zj) * zk;   // for tid<64: j = tid>>3, k = tid&7
        if (tid < DIMP) n1 = c1own + zk;                        // for tid<8:  k = tid

        __syncthreads();
        // ---- write phase ----
        if (tid < D2)   s2[tid] = n2;
        if (tid < DIMP) s1[tid] = n1;
        if (phase == 1) {
            float* o = dst + (size_t)(t0 + s) * SIG;
            float4 q0 = make_float4(c4r[0], c4r[1], c4r[2], c4r[3]);
            float4 q1 = make_float4(c4r[4], c4r[5], c4r[6], c4r[7]);
            *(float4*)(o + OFF4 + tid * 8)     = q0;   // 16B aligned: 4680*4, 584*4, tid*32
            *(float4*)(o + OFF4 + tid * 8 + 4) = q1;
            o[OFF3 + tid] = c3r;
            if (tid < D2)   o[OFF2 + tid] = n2;
            if (tid < DIMP) o[OFF1 + tid] = n1;
        }
        __syncthreads();
    }

    if (phase == 0) {
        float* T = dst + (size_t)c * SIG;
        float4 q0 = make_float4(c4r[0], c4r[1], c4r[2], c4r[3]);
        float4 q1 = make_float4(c4r[4], c4r[5], c4r[6], c4r[7]);
        *(float4*)(T + OFF4 + tid * 8)     = q0;
        *(float4*)(T + OFF4 + tid * 8 + 4) = q1;
        T[OFF3 + tid] = c3r;
        if (tid < D2)   T[OFF2 + tid] = s2[tid];
        if (tid < DIMP) T[OFF1 + tid] = s1[tid];
    }
}

// ---------------------------------------------------------------------------
// Phase 2: sequential scan of chunk totals with the general Chen product.
// Level-4 cross terms a1(x)b3 + a2(x)b2 + a3(x)b1, with c4 viewed as a 64x64
// matrix M[p][q] (p = ij, q = kl), are exactly M += A(64x20) * B(20x64):
//   t in [0,8):  A[p][t] = (t == p&7) ? a1[p>>3] : 0 ; B[t][q] = b3[t*64+q]
//   t in [8,16): A[p][t] = a3[p*8 + t-8]            ; B[t][q] = (t-8 == q>>3) ? b1[q&7] : 0
//   t == 16:     A[p][t] = a2[p]                    ; B[t][q] = b2[q]
//   t in [17,20): zero padding
// Computed with V_WMMA_F32_16X16X4_F32: 16 waves = 16 D-tiles, 5 K-steps each.
// The K-step index is compile-time, so each step's operand build is branch-
// free: unconditional LDS loads + v_cndmask structural selects.
// ---------------------------------------------------------------------------
__global__ __launch_bounds__(TPB)
void sig_phase2_kernel(const float* __restrict__ wsT,
                       float* __restrict__ wsP,
                       int nchunks)
{
    __shared__ float P[SIG];
    __shared__ float Tst[2][SIG];   // double-buffered chunk-total stage
    const int tid = threadIdx.x;

#pragma unroll
    for (int u = 0; u < 10; ++u) {
        const int idx = tid + u * TPB;
        if (idx < SIG) P[idx] = 0.f;
    }
#ifdef ASYNC_LDS
#pragma unroll
    for (int u = 0; u < 3; ++u) {
        const int q4 = tid + u * TPB;
        if (q4 < SIGQ) async_ld_b128(wsT + (size_t)q4 * 4, &Tst[0][q4 * 4]);
    }
    async_wait0();
#else
#pragma unroll
    for (int u = 0; u < 10; ++u) {
        const int idx = tid + u * TPB;
        if (idx < SIG) Tst[0][idx] = wsT[idx];
    }
#endif
    __syncthreads();

    // wave -> 16x16 D-tile mapping for the 64x64 level-4 matrix
    const int w    = tid >> 5;
    const int ln   = tid & 31;
    const int half = ln >> 4;       // K-halves for A/B operands (ISA f32 16x16x4 layout)
    const int lr   = ln & 15;
    const int tr   = w >> 2, tc = w & 3;
    const int p    = tr * 16 + lr;  // A row for this lane
    const int q    = tc * 16 + lr;  // B col for this lane
    const int pk   = p & 7;         // structural-select keys
    const int qh   = q >> 3;

    for (int c = 0; c < nchunks; ++c) {
        const float* Tb = Tst[c & 1];
        float*       Tn = (float*)Tst[(c + 1) & 1];
        const float* a1 = P + OFF1;  const float* a2 = P + OFF2;
        const float* a3 = P + OFF3;  const float* a4 = P + OFF4;
        const float* b1 = Tb + OFF1; const float* b2 = Tb + OFF2;
        const float* b3 = Tb + OFF3; const float* b4 = Tb + OFF4;
        const bool hn = (c + 1 < nchunks);

        // emit exclusive prefix P_c (before applying T_c)
#pragma unroll
        for (int u = 0; u < 10; ++u) {
            const int idx = tid + u * TPB;
            if (idx < SIG) wsP[(size_t)c * SIG + idx] = P[idx];
        }

        // software-pipelined prefetch of T_{c+1} into the other stage buffer
#ifdef ASYNC_LDS
        if (hn) {
#pragma unroll
            for (int u = 0; u < 3; ++u) {
                const int q4 = tid + u * TPB;
                if (q4 < SIGQ)
                    async_ld_b128(wsT + (size_t)(c + 1) * SIG + q4 * 4, Tn + q4 * 4);
            }
        }
#else
        float pf[10];
#pragma unroll
        for (int u = 0; u < 10; ++u) {
            const int idx = tid + u * TPB;
            pf[u] = (hn && idx < SIG) ? wsT[(size_t)(c + 1) * SIG + idx] : 0.f;
        }
#endif

        // levels 1..3 (VALU, one element per thread for level 3)
        const float n3 = a3[tid] + b3[tid]
                       + a1[tid >> 6] * b2[tid & 63]
                       + a2[tid >> 3] * b1[tid & 7];
        float n2 = 0.f, n1 = 0.f;
        if (tid < D2)   n2 = a2[tid] + b2[tid] + a1[tid >> 3] * b1[tid & 7];
        if (tid < DIMP) n1 = a1[tid] + b1[tid];

        // level 4: C-tile init = a4 + b4, then 5 chained WMMA K-steps
        const float a1v = a1[p >> 3];
        const float a2v = a2[p];
        const float b1q = b1[q & 7];
        const float b2v = b2[q];
        v8f acc;
#pragma unroll
        for (int r = 0; r < 8; ++r) {
            const int M  = r + half * 8;
            const int n4 = (tr * 16 + M) * 64 + tc * 16 + lr;
            acc[r] = a4[n4] + b4[n4];
        }
#if __has_builtin(__builtin_amdgcn_wmma_f32_16x16x4_f32)
#pragma unroll
        for (int s = 0; s < 5; ++s) {
            v2f av, bv;
            if (s < 2) {                          // t = 4s + 2*half + {0,1} in [0,8)
                const int ta = 4 * s + 2 * half;
                av.x = (ta     == pk) ? a1v : 0.f;
                av.y = (ta + 1 == pk) ? a1v : 0.f;
                bv.x = b3[ta * 64 + q];
                bv.y = b3[(ta + 1) * 64 + q];
            } else if (s < 4) {                   // t in [8,16), ta = t-8 in [0,8)
                const int ta = 4 * s - 8 + 2 * half;
                av.x = a3[p * 8 + ta];
                av.y = a3[p * 8 + ta + 1];
                bv.x = (ta     == qh) ? b1q : 0.f;
                bv.y = (ta + 1 == qh) ? b1q : 0.f;
            } else {                              // t = 16 (a2*b2), 17..19 zero pad
                av.x = half ? 0.f : a2v;
                av.y = 0.f;
                bv.x = half ? 0.f : b2v;
                bv.y = 0.f;
            }
            acc = __builtin_amdgcn_wmma_f32_16x16x4_f32(false, av, false, bv,
                                                        (short)0, acc, false, false);
        }
#else
        // VALU fallback (also what the host pass parses)
#pragma unroll
        for (int r = 0; r < 8; ++r) {
            const int M  = r + half * 8;
            const int n4 = (tr * 16 + M) * 64 + tc * 16 + lr;
            acc[r] += a1[n4 >> 9] * b3[n4 & 511]
                    + a2[n4 >> 6] * b2[n4 & 63]
                    + a3[n4 >> 3] * b1[n4 & 7];
        }
#endif
        __syncthreads();
        // commit new running product P_{c+1}
        P[OFF3 + tid] = n3;
        if (tid < D2)   P[OFF2 + tid] = n2;
        if (tid < DIMP) P[OFF1 + tid] = n1;
#pragma unroll
        for (int r = 0; r < 8; ++r) {
            const int M  = r + half * 8;
            const int n4 = (tr * 16 + M) * 64 + tc * 16 + lr;
            P[OFF4 + n4] = acc[r];
        }
#ifdef ASYNC_LDS
        async_wait0();            // T_{c+1} DMA landed before everyone passes the barrier
#else
#pragma unroll
        for (int u = 0; u < 10; ++u) {
            const int idx = tid + u * TPB;
            if (hn && idx < SIG) Tn[idx] = pf[u];
        }
#endif
        __syncthreads();
    }
}

// ---------------------------------------------------------------------------
extern "C" void kernel_launch(void* const* d_in, const int* in_sizes, int n_in,
                              void* d_out, int out_size, void* d_ws, size_t ws_size,
                              hipStream_t stream)
{
    (void)n_in; (void)out_size; (void)ws_size;
    const float* path = (const float*)d_in[0];
    float* out = (float*)d_out;

    const int npath   = in_sizes[0] / DIMP;              // 16384
    const int nsteps  = npath - 1;                       // 16383 output rows
    const int nchunks = (nsteps + CLEN - 1) / CLEN;      // 256

    float* wsT = (float*)d_ws;                           // [nchunks][SIG] chunk totals
    float* wsP = wsT + (size_t)nchunks * SIG;            // [nchunks][SIG] exclusive prefixes

    sig_chunk_kernel<<<nchunks, TPB, 0, stream>>>(path, nullptr, wsT, nsteps, 0);
    sig_phase2_kernel<<<1, TPB, 0, stream>>>(wsT, wsP, nchunks);
    sig_chunk_kernel<<<nchunks, TPB, 0, stream>>>(path, wsP, out, nsteps, 1);
}